// EvoformerBlock_42743514530151
// MI455X (gfx1250) — compile-verified
//
#include <hip/hip_runtime.h>
#include <cstdint>

// ---------------------------------------------------------------------------
// Evoformer block for MI455X (gfx1250, wave32, WMMA).
// All matmuls use V_WMMA_F32_16X16X4_F32 (full f32 precision matrix pipe).
// Fragment layouts per CDNA5 ISA 7.12.2:
//   A (16x4): lanes0-15 M=0..15 {K=k0,k0+1}, lanes16-31 M=0..15 {K=k0+2,k0+3}
//   B (4x16): VGPR0 rows K=k0 (lanes0-15) / K=k0+2 (lanes16-31), VGPR1 K+1
//   C/D (16x16 f32): VGPR t: lanes0-15 -> (M=t,N=lane), lanes16-31 -> (M=t+8)
// ---------------------------------------------------------------------------

typedef __attribute__((ext_vector_type(2))) float v2f;
typedef __attribute__((ext_vector_type(8))) float v8f;

__device__ __forceinline__ v8f wmma4(v2f a, v2f b, v8f c) {
  return __builtin_amdgcn_wmma_f32_16x16x4_f32(false, a, false, b, (short)0, c,
                                               false, false);
}

__device__ __forceinline__ float wave_sum(float v) {
#pragma unroll
  for (int m = 16; m > 0; m >>= 1) v += __shfl_xor(v, m, 32);
  return v;
}
__device__ __forceinline__ float wave_max(float v) {
#pragma unroll
  for (int m = 16; m > 0; m >>= 1) v = fmaxf(v, __shfl_xor(v, m, 32));
  return v;
}
__device__ __forceinline__ float sigmoidf(float x) {
  return 1.f / (1.f + __expf(-x));
}

#define GF_ACC 1
#define GF_SIGMOID 2

// ---------------- LayerNorm: 1 wave per row ----------------
__global__ void k_layernorm(const float* __restrict__ x,
                            const float* __restrict__ g,
                            const float* __restrict__ b, float* __restrict__ y,
                            int C) {
  long row = blockIdx.x;
  const float* xr = x + row * C;
  float* yr = y + row * C;
  int lane = threadIdx.x;
  int n = C >> 5;
  float v[4];
  float s = 0.f, s2 = 0.f;
  for (int j = 0; j < n; j++) {
    float t = xr[lane + 32 * j];
    v[j] = t;
    s += t;
    s2 += t * t;
  }
  s = wave_sum(s);
  s2 = wave_sum(s2);
  float mean = s / C;
  float var = s2 / C - mean * mean;
  float rstd = rsqrtf(var + 1e-5f);
  for (int j = 0; j < n; j++) {
    int c = lane + 32 * j;
    yr[c] = (v[j] - mean) * rstd * g[c] + b[c];
  }
}

// ---------------- Generic GEMM: wave computes 16 x (NT*16) tile -------------
// Branch-free K-loop: column indices clamped once up front; invalid output
// columns compute garbage against their own clamped B column and are simply
// never stored. Keeps the 8-VGPR accumulators pinned (no CFG merges around
// v_wmma), which avoids the PHI-copy storm seen with in-loop guards.
template <int NT, bool GUARD>
__device__ __forceinline__ void gemm_body(const float* __restrict__ A,
                                          const float* __restrict__ B,
                                          const float* __restrict__ bias,
                                          float* __restrict__ C, int N, int K,
                                          int ldb, int ldc, int flags,
                                          int nbase) {
  int lane = threadIdx.x;
  int hh = lane >> 4, r = lane & 15;
  long m0 = (long)blockIdx.y * 16;
  v8f acc[NT] = {};
  const float* Arow = A + (m0 + r) * (long)K;
  int cols[NT];
#pragma unroll
  for (int t = 0; t < NT; t++) {
    int col = nbase + t * 16 + r;
    cols[t] = (!GUARD || col < N) ? col : 0;  // clamp, resolved pre-loop
  }
  for (int kk = 0; kk < K; kk += 4) {
    int k0 = kk + 2 * hh;
    v2f a;
    a.x = Arow[k0];
    a.y = Arow[k0 + 1];
    const float* B0 = B + (long)k0 * ldb;
    const float* B1 = B0 + ldb;
#pragma unroll
    for (int t = 0; t < NT; t++) {
      v2f bf;
      bf.x = B0[cols[t]];
      bf.y = B1[cols[t]];
      acc[t] = wmma4(a, bf, acc[t]);
    }
  }
#pragma unroll
  for (int t = 0; t < NT; t++) {
    int col = nbase + t * 16 + r;
    if (GUARD && col >= N) continue;
#pragma unroll
    for (int tt = 0; tt < 8; tt++) {
      long row = m0 + tt + 8 * hh;
      float val = acc[t][tt];
      if (bias) val += bias[col];
      if (flags & GF_SIGMOID) val = sigmoidf(val);
      float* cp = C + row * ldc + col;
      if (flags & GF_ACC)
        *cp += val;
      else
        *cp = val;
    }
  }
}

// C[M,N] = op(A[M,K] @ B[K,N] (+bias)); A row-major lda=K, B ldb, C ldc.
// Requires M % 16 == 0, K % 4 == 0. N arbitrary.
__global__ void k_gemm(const float* __restrict__ A, const float* __restrict__ B,
                       const float* __restrict__ bias, float* __restrict__ C,
                       int M, int N, int K, int ldb, int ldc, int flags) {
  int nbase = blockIdx.x * 64;
  int rem = N - nbase;
  if (rem >= 64)
    gemm_body<4, false>(A, B, bias, C, N, K, ldb, ldc, flags, nbase);
  else if (rem == 48)
    gemm_body<3, false>(A, B, bias, C, N, K, ldb, ldc, flags, nbase);
  else if (rem > 48)
    gemm_body<4, true>(A, B, bias, C, N, K, ldb, ldc, flags, nbase);
  else if (rem == 32)
    gemm_body<2, false>(A, B, bias, C, N, K, ldb, ldc, flags, nbase);
  else if (rem > 32)
    gemm_body<3, true>(A, B, bias, C, N, K, ldb, ldc, flags, nbase);
  else if (rem == 16)
    gemm_body<1, false>(A, B, bias, C, N, K, ldb, ldc, flags, nbase);
  else if (rem > 16)
    gemm_body<2, true>(A, B, bias, C, N, K, ldb, ldc, flags, nbase);
  else
    gemm_body<1, true>(A, B, bias, C, N, K, ldb, ldc, flags, nbase);
}

// ---------------- Fused transition: X(16 rows) -> GLU(LDS) -> +=Y ----------
// W1: (C, 8C)  [a half cols 0..4C, b half cols 4C..8C], W2: (4C, C), H = 4C.
__global__ void k_transition(const float* __restrict__ X,
                             const float* __restrict__ W1,
                             const float* __restrict__ W2,
                             float* __restrict__ Y, int C) {
  __shared__ float glu[16 * 513];  // max: H=512 (+1 pad)
  int H = 4 * C;
  int gld = H + 1;
  int ld1 = 8 * C;
  int tid = threadIdx.x;
  int wave = tid >> 5;
  int lane = tid & 31;
  int hh = lane >> 4, r = lane & 15;
  long m0 = (long)blockIdx.x * 16;
  const float* Xr = X + (m0 + r) * (long)C;
  for (int jt = wave; jt < H / 16; jt += 4) {
    v8f aa = {}, bb = {};
    int col = jt * 16 + r;
    for (int kk = 0; kk < C; kk += 4) {
      int k0 = kk + 2 * hh;
      v2f a;
      a.x = Xr[k0];
      a.y = Xr[k0 + 1];
      v2f b1, b2;
      b1.x = W1[(long)k0 * ld1 + col];
      b1.y = W1[(long)(k0 + 1) * ld1 + col];
      b2.x = W1[(long)k0 * ld1 + H + col];
      b2.y = W1[(long)(k0 + 1) * ld1 + H + col];
      aa = wmma4(a, b1, aa);
      bb = wmma4(a, b2, bb);
    }
#pragma unroll
    for (int tt = 0; tt < 8; tt++) {
      int m = tt + 8 * hh;
      float av = aa[tt];
      float sl = av * sigmoidf(av);  // silu
      glu[m * gld + jt * 16 + r] = sl * bb[tt];
    }
  }
  __syncthreads();
  for (int nt = wave; nt < C / 16; nt += 4) {
    v8f acc = {};
    int col = nt * 16 + r;
    for (int kk = 0; kk < H; kk += 4) {
      int k0 = kk + 2 * hh;
      v2f a;
      a.x = glu[r * gld + k0];
      a.y = glu[r * gld + k0 + 1];
      v2f b;
      b.x = W2[(long)k0 * C + col];
      b.y = W2[(long)(k0 + 1) * C + col];
      acc = wmma4(a, b, acc);
    }
#pragma unroll
    for (int tt = 0; tt < 8; tt++) {
      long row = m0 + tt + 8 * hh;
      Y[row * C + col] += acc[tt];
    }
  }
}

// ---------------- OPM fused block: O = L(:,i)^T R(:,j); out = O:W + b -------
// L,R: (512, 256, 32). W: (32,32,128). norm: (256,256). pair += out/(eps+norm)
__global__ void k_opm(const float* __restrict__ L, const float* __restrict__ R,
                      const float* __restrict__ W, const float* __restrict__ bias,
                      const float* __restrict__ norm, float* __restrict__ pair) {
  __shared__ float O[32 * 32];
  int i = blockIdx.y, j = blockIdx.x;
  int tid = threadIdx.x;
  int wave = tid >> 5, lane = tid & 31;
  int hh = lane >> 4, r = lane & 15;
  int c0 = (wave & 1) * 16, e0 = (wave >> 1) * 16;
  v8f acc = {};
  for (int kk = 0; kk < 512; kk += 4) {
    int k0 = kk + 2 * hh;
    v2f a;
    a.x = L[((long)k0 * 256 + i) * 32 + c0 + r];
    a.y = L[((long)(k0 + 1) * 256 + i) * 32 + c0 + r];
    v2f b;
    b.x = R[((long)k0 * 256 + j) * 32 + e0 + r];
    b.y = R[((long)(k0 + 1) * 256 + j) * 32 + e0 + r];
    acc = wmma4(a, b, acc);
  }
#pragma unroll
  for (int tt = 0; tt < 8; tt++) O[(c0 + tt + 8 * hh) * 32 + e0 + r] = acc[tt];
  __syncthreads();
  int f = tid;  // 0..127
  float s = 0.f;
  for (int ce = 0; ce < 1024; ce++) s += O[ce] * W[(long)ce * 128 + f];
  float out = (s + bias[f]) / (0.001f + norm[i * 256 + j]);
  pair[((long)i * 256 + j) * 128 + f] += out;
}

// ---------------- Triangle mult per-channel GEMM ----------------
// TA/TB: channel-major (128, 256, 256). outgoing: out=X Y^T ; else X^T Y.
// out written to (i,j,c) layout.
__global__ void k_tri(const float* __restrict__ TA, const float* __restrict__ TB,
                      float* __restrict__ out, int outgoing) {
  int c = blockIdx.z;
  int j0 = blockIdx.x * 16, i0 = blockIdx.y * 16;
  const float* X = TA + (long)c * 65536;
  const float* Y = TB + (long)c * 65536;
  int lane = threadIdx.x, hh = lane >> 4, r = lane & 15;
  v8f acc = {};
  if (outgoing) {
    for (int kk = 0; kk < 256; kk += 4) {
      int k0 = kk + 2 * hh;
      v2f a;
      a.x = X[(i0 + r) * 256 + k0];
      a.y = X[(i0 + r) * 256 + k0 + 1];
      v2f b;
      b.x = Y[(j0 + r) * 256 + k0];
      b.y = Y[(j0 + r) * 256 + k0 + 1];
      acc = wmma4(a, b, acc);
    }
  } else {
    for (int kk = 0; kk < 256; kk += 4) {
      int k0 = kk + 2 * hh;
      v2f a;
      a.x = X[k0 * 256 + i0 + r];
      a.y = X[(k0 + 1) * 256 + i0 + r];
      v2f b;
      b.x = Y[k0 * 256 + j0 + r];
      b.y = Y[(k0 + 1) * 256 + j0 + r];
      acc = wmma4(a, b, acc);
    }
  }
#pragma unroll
  for (int tt = 0; tt < 8; tt++)
    out[((long)(i0 + tt + 8 * hh) * 256 + j0 + r) * 128 + c] = acc[tt];
}

// ---------------- Pair grid attention block (one head, 16 q rows) ----------
// Q/K/V: per-head (65536, 32). bias: (q,k,4). O: (65536,128) at col h*32.
__global__ void k_pair_attn(const float* __restrict__ Q,
                            const float* __restrict__ Kt,
                            const float* __restrict__ V,
                            const float* __restrict__ bias,
                            const float* __restrict__ mask,
                            float* __restrict__ O, int h) {
  __shared__ float lw[16 * 257];
  int q0 = blockIdx.x * 16;
  int b = blockIdx.y;
  int lane = threadIdx.x, hh = lane >> 4, r = lane & 15;
  const float scale = 0.17677669529663687f;  // 32^-0.5
  long base = (long)b * 256;
  for (int nt = 0; nt < 16; nt++) {
    v8f acc = {};
    for (int kk = 0; kk < 32; kk += 4) {
      int k0 = kk + 2 * hh;
      v2f a;
      a.x = Q[(base + q0 + r) * 32 + k0];
      a.y = Q[(base + q0 + r) * 32 + k0 + 1];
      v2f bf;
      bf.x = Kt[(base + nt * 16 + r) * 32 + k0];
      bf.y = Kt[(base + nt * 16 + r) * 32 + k0 + 1];
      acc = wmma4(a, bf, acc);
    }
#pragma unroll
    for (int tt = 0; tt < 8; tt++) {
      int m = tt + 8 * hh;
      int kcol = nt * 16 + r;
      lw[m * 257 + kcol] = acc[tt] * scale +
                           bias[((long)(q0 + m) * 256 + kcol) * 4 + h] +
                           1e9f * (mask[base + kcol] - 1.f);
    }
  }
  __syncthreads();
  for (int m = 0; m < 16; m++) {
    float mx = -3.4e38f;
    for (int kc = lane; kc < 256; kc += 32) mx = fmaxf(mx, lw[m * 257 + kc]);
    mx = wave_max(mx);
    float sum = 0.f;
    for (int kc = lane; kc < 256; kc += 32) {
      float e = __expf(lw[m * 257 + kc] - mx);
      lw[m * 257 + kc] = e;
      sum += e;
    }
    sum = wave_sum(sum);
    float inv = 1.f / sum;
    for (int kc = lane; kc < 256; kc += 32) lw[m * 257 + kc] *= inv;
  }
  __syncthreads();
  for (int nt = 0; nt < 2; nt++) {
    v8f acc = {};
    for (int kk = 0; kk < 256; kk += 4) {
      int k0 = kk + 2 * hh;
      v2f a;
      a.x = lw[r * 257 + k0];
      a.y = lw[r * 257 + k0 + 1];
      v2f bf;
      bf.x = V[(base + k0) * 32 + nt * 16 + r];
      bf.y = V[(base + k0 + 1) * 32 + nt * 16 + r];
      acc = wmma4(a, bf, acc);
    }
#pragma unroll
    for (int tt = 0; tt < 8; tt++)
      O[(base + q0 + tt + 8 * hh) * 128 + h * 32 + nt * 16 + r] = acc[tt];
  }
}

// ---------------- MSA attention: o[s,q,h*8+d] = sum_k w[h,q,k] v[s,k,h*8+d]
__global__ void k_msa_attn_o(const float* __restrict__ Wb,
                             const float* __restrict__ V,
                             float* __restrict__ O) {
  int q0 = blockIdx.x * 16;
  int h = blockIdx.y;
  int s = blockIdx.z;
  int lane = threadIdx.x, hh = lane >> 4, r = lane & 15;
  v8f acc = {};
  const float* wh = Wb + (long)h * 65536;
  long sb = (long)s * 256;
  int vc = h * 8 + ((r < 8) ? r : 0);  // clamped column (unstored when r>=8)
  for (int kk = 0; kk < 256; kk += 4) {
    int k0 = kk + 2 * hh;
    v2f a;
    a.x = wh[(q0 + r) * 256 + k0];
    a.y = wh[(q0 + r) * 256 + k0 + 1];
    v2f b;
    b.x = V[(sb + k0) * 64 + vc];
    b.y = V[(sb + k0 + 1) * 64 + vc];
    acc = wmma4(a, b, acc);
  }
  if (r < 8) {
#pragma unroll
    for (int tt = 0; tt < 8; tt++)
      O[(sb + q0 + tt + 8 * hh) * 64 + h * 8 + r] = acc[tt];
  }
}

// ---------------- MSA attention softmax over k (row = (h,q)) ----------------
__global__ void k_msa_softmax(const float* __restrict__ logits,
                              const float* __restrict__ colmax,
                              float* __restrict__ Wb) {
  int q = blockIdx.x, h = blockIdx.y;
  int lane = threadIdx.x;
  float vals[8];
  float mx = -3.4e38f;
  for (int t = 0; t < 8; t++) {
    int k = lane + 32 * t;
    float v = logits[((long)q * 256 + k) * 8 + h] + 1e9f * (colmax[k] - 1.f);
    vals[t] = v;
    mx = fmaxf(mx, v);
  }
  mx = wave_max(mx);
  float sum = 0.f;
  for (int t = 0; t < 8; t++) {
    vals[t] = __expf(vals[t] - mx);
    sum += vals[t];
  }
  sum = wave_sum(sum);
  float inv = 1.f / sum;
  for (int t = 0; t < 8; t++)
    Wb[(long)h * 65536 + q * 256 + lane + 32 * t] = vals[t] * inv;
}

// ---------------- small helpers ----------------
__global__ void k_norm_ij(const float* __restrict__ m, float* __restrict__ nrm) {
  int i = blockIdx.x, j = threadIdx.x;
  float s = 0.f;
  for (int sx = 0; sx < 512; sx++) s += m[sx * 256 + i] * m[sx * 256 + j];
  nrm[i * 256 + j] = s;
}
__global__ void k_colmax(const float* __restrict__ m, float* __restrict__ cm) {
  int k = threadIdx.x;
  float mx = -3.4e38f;
  for (int s = 0; s < 512; s++) mx = fmaxf(mx, m[s * 256 + k]);
  cm[k] = mx;
}
__global__ void k_rowscale(float* __restrict__ a, const float* __restrict__ s,
                           int cols, long n) {
  long i = (long)blockIdx.x * 256 + threadIdx.x;
  if (i < n) a[i] *= s[i / cols];
}
__global__ void k_projgate(float* __restrict__ p, const float* __restrict__ g,
                           const float* __restrict__ mask, long n) {
  long i = (long)blockIdx.x * 256 + threadIdx.x;
  if (i < n) p[i] *= mask[i / 256] * g[i];
}
__global__ void k_mul(float* __restrict__ a, const float* __restrict__ b, long n) {
  long i = (long)blockIdx.x * 256 + threadIdx.x;
  if (i < n) a[i] *= b[i];
}
__global__ void k_muladd(float* __restrict__ d, const float* __restrict__ a,
                         const float* __restrict__ b, long n) {
  long i = (long)blockIdx.x * 256 + threadIdx.x;
  if (i < n) d[i] += a[i] * b[i];
}
__global__ void k_swap01(const float* __restrict__ in, float* __restrict__ out) {
  int i = blockIdx.y, j = blockIdx.x, c = threadIdx.x;
  out[((long)j * 256 + i) * 128 + c] = in[((long)i * 256 + j) * 128 + c];
}
__global__ void k_addT(float* __restrict__ dst, const float* __restrict__ src) {
  int i = blockIdx.y, j = blockIdx.x, c = threadIdx.x;
  dst[((long)i * 256 + j) * 128 + c] += src[((long)j * 256 + i) * 128 + c];
}
// out[c*rows + p] = in[p*ldin + coff + c]; tile 32x32 via LDS.
__global__ void k_transpose32(const float* __restrict__ in,
                              float* __restrict__ out, int rows, int ldin,
                              int coff) {
  __shared__ float t[32][33];
  int p0 = blockIdx.x * 32, c0 = blockIdx.y * 32;
  int tx = threadIdx.x, ty = threadIdx.y;  // (32,8)
  for (int yy = 0; yy < 4; yy++) {
    int pl = ty + yy * 8;
    t[pl][tx] = in[(long)(p0 + pl) * ldin + coff + c0 + tx];
  }
  __syncthreads();
  for (int yy = 0; yy < 4; yy++) {
    int cl = ty + yy * 8;
    out[(long)(c0 + cl) * rows + p0 + tx] = t[tx][cl];
  }
}

// ---------------------------------------------------------------------------
// Host orchestration
// ---------------------------------------------------------------------------
static inline void ln_launch(hipStream_t st, const float* x, const float* g,
                             const float* b, float* y, long rows, int C) {
  k_layernorm<<<dim3((unsigned)rows), dim3(32), 0, st>>>(x, g, b, y, C);
}
static inline void gemm_launch(hipStream_t st, const float* A, const float* B,
                               const float* bias, float* C, long M, int N,
                               int K, int ldb, int ldc, int flags) {
  k_gemm<<<dim3((N + 63) / 64, (unsigned)(M / 16)), dim3(32), 0, st>>>(
      A, B, bias, C, (int)M, N, K, ldb, ldc, flags);
}
static inline unsigned ewb(long n) { return (unsigned)((n + 255) / 256); }

extern "C" void kernel_launch(void* const* d_in, const int* in_sizes, int n_in,
                              void* d_out, int out_size, void* d_ws,
                              size_t ws_size, hipStream_t stream) {
  (void)in_sizes; (void)n_in; (void)out_size; (void)ws_size;
  const long SN = 512L * 256;  // msa rows
  const long NN = 256L * 256;  // pair rows
  const int CM = 64, CP = 128;

  const float* msa_in = (const float*)d_in[0];
  const float* pair_in = (const float*)d_in[1];
  const float* msa_mask = (const float*)d_in[2];
  const float* pair_mask = (const float*)d_in[3];
  const float* mask_attn = (const float*)d_in[4];
  const float* P[54];
  for (int k = 0; k < 54; k++) P[k] = (const float*)d_in[5 + k];

  float* msa = (float*)d_out;           // (512,256,64)
  float* pair = msa + SN * CM;          // (256,256,128)

  float* ws = (float*)d_ws;
  float* Ab = ws;                        //  8.39M floats
  float* Bb = Ab + 8388608L;             // 16.78M
  float* Cc = Bb + 16777216L;            // 16.78M
  float* Dd = Cc + 16777216L;            // 16.78M
  float* Ee = Dd + 16777216L;            //  8.39M
  float* Ff = Ee + 8388608L;             //  8.39M
  float* Gg = Ff + 8388608L;             //  8.39M

  hipMemcpyAsync(msa, msa_in, SN * CM * sizeof(float),
                 hipMemcpyDeviceToDevice, stream);
  hipMemcpyAsync(pair, pair_in, NN * CP * sizeof(float),
                 hipMemcpyDeviceToDevice, stream);

  // ======== 1) pair += outer_product_mean(msa) ========
  ln_launch(stream, msa, P[0], P[1], Ab, SN, CM);
  float* left = Dd;
  float* right = Dd + 4194304L;
  gemm_launch(stream, Ab, P[2], nullptr, left, SN, 32, 64, 32, 32, 0);
  gemm_launch(stream, Ab, P[3], nullptr, right, SN, 32, 64, 32, 32, 0);
  k_rowscale<<<ewb(SN * 32), 256, 0, stream>>>(left, msa_mask, 32, SN * 32);
  k_rowscale<<<ewb(SN * 32), 256, 0, stream>>>(right, msa_mask, 32, SN * 32);
  float* nrm = Bb;
  k_norm_ij<<<dim3(256), 256, 0, stream>>>(msa_mask, nrm);
  k_opm<<<dim3(256, 256), 128, 0, stream>>>(left, right, P[4], P[5], nrm, pair);

  // ======== 2) msa += msa_attention(msa, pair) ========
  ln_launch(stream, msa, P[6], P[7], Ab, SN, CM);    // x
  ln_launch(stream, pair, P[8], P[9], Ff, NN, CP);   // z
  float* logitsm = Bb + 65536L;
  gemm_launch(stream, Ff, P[10], nullptr, logitsm, NN, 8, 128, 8, 8, 0);
  float* colmax = Bb + 65536L + 524288L;
  k_colmax<<<1, 256, 0, stream>>>(msa_mask, colmax);
  float* wbuf = Cc;  // (8,256,256)
  k_msa_softmax<<<dim3(256, 8), 32, 0, stream>>>(logitsm, colmax, wbuf);
  gemm_launch(stream, Ab, P[11], nullptr, Ee, SN, 64, 64, 64, 64, 0);  // v
  k_msa_attn_o<<<dim3(16, 8, 512), 32, 0, stream>>>(wbuf, Ee, Ff);     // o
  gemm_launch(stream, Ab, P[12], nullptr, Gg, SN, 64, 64, 64, 64, GF_SIGMOID);
  k_mul<<<ewb(SN * 64), 256, 0, stream>>>(Ff, Gg, SN * 64);
  gemm_launch(stream, Ff, P[13], nullptr, msa, SN, 64, 64, 64, 64, GF_ACC);

  // ======== 3) msa += transition(msa) ========
  ln_launch(stream, msa, P[14], P[15], Ab, SN, CM);
  k_transition<<<dim3((unsigned)(SN / 16)), 128, 0, stream>>>(Ab, P[16], P[17],
                                                              msa, CM);

  // ======== 4/5) pair += triangle_mult (outgoing, incoming) ========
  for (int t = 0; t < 2; t++) {
    int pb = (t == 0) ? 18 : 26;
    int outgoing = (t == 0) ? 1 : 0;
    ln_launch(stream, pair, P[pb], P[pb + 1], Ab, NN, CP);
    gemm_launch(stream, Ab, P[pb + 2], nullptr, Bb, NN, 256, 128, 256, 256, 0);
    gemm_launch(stream, Ab, P[pb + 3], nullptr, Cc, NN, 256, 128, 256, 256,
                GF_SIGMOID);
    k_projgate<<<ewb(NN * 256), 256, 0, stream>>>(Bb, Cc, pair_mask, NN * 256);
    k_transpose32<<<dim3(2048, 4), dim3(32, 8), 0, stream>>>(Bb, Dd, (int)NN,
                                                             256, 0);
    k_transpose32<<<dim3(2048, 4), dim3(32, 8), 0, stream>>>(
        Bb, Dd + 8388608L, (int)NN, 256, 128);
    k_tri<<<dim3(16, 16, 128), 32, 0, stream>>>(Dd, Dd + 8388608L, Ee,
                                                outgoing);
    ln_launch(stream, Ee, P[pb + 4], P[pb + 5], Ff, NN, CP);
    gemm_launch(stream, Ff, P[pb + 6], nullptr, Gg, NN, 128, 128, 128, 128, 0);
    gemm_launch(stream, Ab, P[pb + 7], nullptr, Ee, NN, 128, 128, 128, 128,
                GF_SIGMOID);
    k_muladd<<<ewb(NN * 128), 256, 0, stream>>>(pair, Gg, Ee, NN * 128);
  }

  // ======== 6/7) pair += grid_attention (rows, cols-transposed) ========
  for (int t = 0; t < 2; t++) {
    int pb = (t == 0) ? 34 : 42;
    int tr = t;
    ln_launch(stream, pair, P[pb], P[pb + 1], Ab, NN, CP);
    float* biasb = Bb;  // (q,k,4) from pre-transpose x
    gemm_launch(stream, Ab, P[pb + 2], nullptr, biasb, NN, 4, 128, 4, 4, 0);
    const float* xw = Ab;
    if (tr) {
      k_swap01<<<dim3(256, 256), 128, 0, stream>>>(Ab, Cc);
      xw = Cc;
    }
    gemm_launch(stream, xw, P[pb + 6], nullptr, Ff, NN, 128, 128, 128, 128,
                GF_SIGMOID);  // gq
    float* qh = Dd;
    float* kh = Dd + 2097152L;
    float* vh = Dd + 4194304L;
    for (int h = 0; h < 4; h++) {
      gemm_launch(stream, xw, P[pb + 3] + h * 32, nullptr, qh, NN, 32, 128, 128,
                  32, 0);
      gemm_launch(stream, xw, P[pb + 4] + h * 32, nullptr, kh, NN, 32, 128, 128,
                  32, 0);
      gemm_launch(stream, xw, P[pb + 5] + h * 32, nullptr, vh, NN, 32, 128, 128,
                  32, 0);
      k_pair_attn<<<dim3(16, 256), 32, 0, stream>>>(qh, kh, vh, biasb,
                                                    mask_attn, Ee, h);
    }
    k_mul<<<ewb(NN * 128), 256, 0, stream>>>(Ee, Ff, NN * 128);
    if (!tr) {
      gemm_launch(stream, Ee, P[pb + 7], nullptr, pair, NN, 128, 128, 128, 128,
                  GF_ACC);
    } else {
      gemm_launch(stream, Ee, P[pb + 7], nullptr, Gg, NN, 128, 128, 128, 128,
                  0);
      k_addT<<<dim3(256, 256), 128, 0, stream>>>(pair, Gg);
    }
  }

  // ======== 8) pair += transition(pair) ========
  ln_launch(stream, pair, P[50], P[51], Ab, NN, CP);
  k_transition<<<dim3((unsigned)(NN / 16)), 128, 0, stream>>>(Ab, P[52], P[53],
                                                              pair, CP);
}